// TopKPooling_68685116997716
// MI455X (gfx1250) — compile-verified
//
#include <hip/hip_runtime.h>
#include <hip/hip_bf16.h>

typedef __attribute__((ext_vector_type(2))) float v2f;
typedef __attribute__((ext_vector_type(8))) float v8f;

#define D_FEAT 256

// ---------------------------------------------------------------------------
// ws layout (as u32 words):
//   [0..255]   radix histogram
//   [256]      prefix (becomes threshold key T)
//   [257]      remaining count within current prefix (becomes #eq to keep)
//   [259]      atomic counter: slots for keys >  T
//   [260]      atomic counter: budget for keys == T
//   [512 .. 512+N)        keys   (order-preserving u32 of score)
//   [512+N .. 512+2N)     sel    (membership table, 0/1)
//   [512+2N .. 512+2N+K)  idx    (compacted kept-node list)
// ---------------------------------------------------------------------------

__global__ void tk_init(unsigned* __restrict__ W, int K) {
    int t = blockIdx.x * blockDim.x + threadIdx.x;
    if (t < 512) W[t] = (t == 257) ? (unsigned)K : 0u;
}

// score = x @ w via V_WMMA_F32_16X16X4_F32, one wave per 16-row tile.
// B = w chunk broadcast into all 16 columns, so every column of D holds the
// scores. K is fed in a permuted order so each lane's inputs for TWO wmma
// steps come from one B128 load:
//   step0: lanes0-15 K={k,k+1},   lanes16-31 K={k+4,k+5}
//   step1: lanes0-15 K={k+2,k+3}, lanes16-31 K={k+6,k+7}
// (valid because sum over K is order-independent and A/B use the same map).
__global__ void tk_score(const float* __restrict__ x, const float* __restrict__ w,
                         unsigned* __restrict__ keys, int N) {
    __shared__ __align__(16) float lw[D_FEAT];
    for (int t = threadIdx.x; t < D_FEAT; t += blockDim.x) lw[t] = w[t];
    __syncthreads();

    const int wave = threadIdx.x >> 5;
    const int lane = threadIdx.x & 31;
    const int tile = blockIdx.x * 8 + wave;      // 16 rows per tile
    const int ntiles = N >> 4;                   // N is a multiple of 16
    if (tile >= ntiles) return;

    const int row0 = tile << 4;
    const int arow = row0 + (lane & 15);
    const int hoff = (lane >> 4) << 2;           // 0 (lanes 0-15) or 4 (16-31)
    const float* xp  = x + (size_t)arow * D_FEAT + hoff;
    const float* lwp = lw + hoff;

    v8f c = {};
    #pragma unroll 2
    for (int kb = 0; kb < D_FEAT; kb += 8) {
        const float4 av = *(const float4*)(xp + kb);   // global_load_b128
        const float4 bv = *(const float4*)(lwp + kb);  // ds_load_b128
        v2f a0 = {av.x, av.y}, a1 = {av.z, av.w};
        v2f b0 = {bv.x, bv.y}, b1 = {bv.z, bv.w};
        // 8 args: (neg_a, A, neg_b, B, c_mod, C, reuse_a, reuse_b)
        c = __builtin_amdgcn_wmma_f32_16x16x4_f32(false, a0, false, b0,
                                                  (short)0, c, false, false);
        c = __builtin_amdgcn_wmma_f32_16x16x4_f32(false, a1, false, b1,
                                                  (short)0, c, false, false);
    }

    // C/D layout: VGPR r = row r (lanes 0-15) / row r+8 (lanes 16-31);
    // all 16 columns identical -> lane 0 and lane 16 hold every score.
    if ((lane & 15) == 0) {
        const int rbase = row0 + ((lane >> 4) << 3);
        #pragma unroll
        for (int m = 0; m < 8; ++m) {
            unsigned b = __float_as_uint(c[m]);
            keys[rbase + m] = b ^ ((b >> 31) ? 0xFFFFFFFFu : 0x80000000u);
        }
    }
}

// One radix-select round: histogram the 8-bit digit at `shift` over keys whose
// high bits match the current prefix. LDS histogram, flushed with atomics.
__global__ void tk_hist(const unsigned* __restrict__ keys, unsigned* __restrict__ W,
                        int N, int shift) {
    __shared__ unsigned lh[256];
    for (int t = threadIdx.x; t < 256; t += blockDim.x) lh[t] = 0u;
    __syncthreads();
    const unsigned prefix = W[256];
    int i = blockIdx.x * blockDim.x + threadIdx.x;
    const int stride = gridDim.x * blockDim.x;
    for (; i < N; i += stride) {
        unsigned k = keys[i];
        if (shift < 24 && (k >> (shift + 8)) != prefix) continue;
        atomicAdd(&lh[(k >> shift) & 255u], 1u);
    }
    __syncthreads();
    for (int t = threadIdx.x; t < 256; t += blockDim.x)
        if (lh[t]) atomicAdd(&W[t], lh[t]);
}

// Single block: walk bins 255..0, find the bin containing the remaining-th
// largest, refine prefix/remaining, zero the histogram for the next round.
__global__ void tk_scan(unsigned* __restrict__ W) {
    __shared__ unsigned h[256];
    h[threadIdx.x] = W[threadIdx.x];
    __syncthreads();
    if (threadIdx.x == 0) {
        unsigned remaining = W[257];
        unsigned prefix    = W[256];
        unsigned cum = 0; int bsel = 0;
        for (int b = 255; b >= 0; --b) {
            unsigned nc = cum + h[b];
            if (nc >= remaining) { bsel = b; remaining -= cum; break; }
            cum = nc;
        }
        W[256] = (prefix << 8) | (unsigned)bsel;
        W[257] = remaining;
    }
    __syncthreads();
    W[threadIdx.x] = 0u;
}

// Exact-K selection: all keys > T kept; exactly `rem` keys == T kept.
__global__ void tk_select(const unsigned* __restrict__ keys, unsigned* __restrict__ W,
                          unsigned* __restrict__ sel, int* __restrict__ idx,
                          int N, int K) {
    const unsigned T   = W[256];
    const unsigned rem = W[257];
    const unsigned G   = (unsigned)K - rem;      // # strictly greater than T
    int i = blockIdx.x * blockDim.x + threadIdx.x;
    const int stride = gridDim.x * blockDim.x;
    for (; i < N; i += stride) {
        unsigned k = keys[i];
        unsigned s = 0u;
        if (k > T) {
            unsigned slot = atomicAdd(&W[259], 1u);
            idx[slot] = i; s = 1u;
        } else if (k == T) {
            unsigned e = atomicAdd(&W[260], 1u);
            if (e < rem) { idx[G + e] = i; s = 1u; }
        }
        sel[i] = s;
    }
}

// x_sliced = x[idx] : wave per row, 2x B128 per lane. x is L2-resident (102MB < 192MB).
__global__ void tk_gather(const float* __restrict__ x, const int* __restrict__ idx,
                          float* __restrict__ out, int K) {
    const int wave = threadIdx.x >> 5;
    const int lane = threadIdx.x & 31;
    const int r = blockIdx.x * 8 + wave;
    if (r >= K) return;
    const float4* src = (const float4*)(x + (size_t)idx[r] * D_FEAT);
    float4* dst = (float4*)(out + (size_t)r * D_FEAT);
    dst[lane]      = src[lane];
    dst[lane + 32] = src[lane + 32];
}

// Edge filter: mask = sel[Ai] & sel[Aj]; kept edges keep ids, dropped -> -1.
__global__ void tk_edges(const int* __restrict__ Ai, const int* __restrict__ Aj,
                         const unsigned* __restrict__ sel,
                         int* __restrict__ outAi, int* __restrict__ outAj,
                         int* __restrict__ outM, int E) {
    const int e = blockIdx.x * blockDim.x + threadIdx.x;
    if (e >= E) return;
    const int ai = Ai[e], aj = Aj[e];
    const unsigned m = sel[ai] & sel[aj];
    outAi[e] = m ? ai : -1;
    outAj[e] = m ? aj : -1;
    outM[e]  = (int)m;
}

extern "C" void kernel_launch(void* const* d_in, const int* in_sizes, int n_in,
                              void* d_out, int out_size, void* d_ws, size_t ws_size,
                              hipStream_t stream) {
    const float* x  = (const float*)d_in[0];
    const int*   Ai = (const int*)d_in[1];
    const int*   Aj = (const int*)d_in[2];
    const float* w  = (const float*)d_in[3];

    const int N = in_sizes[0] / D_FEAT;   // 100000
    const int E = in_sizes[1];            // 3200000
    const int K = N / 2;                  // 50000

    unsigned* W    = (unsigned*)d_ws;
    unsigned* keys = W + 512;
    unsigned* sel  = keys + N;
    int*      idx  = (int*)(sel + N);

    float* out   = (float*)d_out;
    int*   outAi = (int*)(out + (size_t)K * D_FEAT);
    int*   outAj = outAi + E;
    int*   outM  = outAj + E;

    tk_init<<<2, 256, 0, stream>>>(W, K);

    const int ntiles = N / 16;
    tk_score<<<(ntiles + 7) / 8, 256, 0, stream>>>(x, w, keys, N);

    for (int shift = 24; shift >= 0; shift -= 8) {
        tk_hist<<<512, 256, 0, stream>>>(keys, W, N, shift);
        tk_scan<<<1, 256, 0, stream>>>(W);
    }

    tk_select<<<512, 256, 0, stream>>>(keys, W, sel, idx, N, K);
    tk_gather<<<(K + 7) / 8, 256, 0, stream>>>(x, idx, out, K);
    tk_edges<<<(E + 255) / 256, 256, 0, stream>>>(Ai, Aj, sel, outAi, outAj, outM, E);
}